// PriorConditionedSSM_36172214567197
// MI455X (gfx1250) — compile-verified
//
#include <hip/hip_runtime.h>
#include <hip/hip_bf16.h>

typedef float v2f __attribute__((ext_vector_type(2)));
typedef float v8f __attribute__((ext_vector_type(8)));

#define Bsz 8
#define Cch 64
#define Dch 128
#define Him 128
#define Wim 128
#define HW  (Him * Wim)                       // 16384
#define BDHW ((size_t)Bsz * Dch * HW)         // 16,777,216 floats

// ---------------------------------------------------------------------------
// K0: tiny precompute.  M = Wo @ Wf  (64 x 512),  c0 = Wo@bf + bo,  At = tanh(A)
// ---------------------------------------------------------------------------
__global__ void k0_precompute(const float* __restrict__ Wo, const float* __restrict__ Wf,
                              const float* __restrict__ bf, const float* __restrict__ bo,
                              const float* __restrict__ A_param,
                              float* __restrict__ Mf, float* __restrict__ c0,
                              float* __restrict__ At) {
    const int tid = threadIdx.x;               // 256 threads, 1 block
    if (tid < Dch) At[tid] = tanhf(A_param[tid]);
    if (tid < Cch) {
        float s = bo[tid];
        for (int d = 0; d < Dch; ++d) s += Wo[tid * Dch + d] * bf[d];
        c0[tid] = s;
    }
    for (int idx = tid; idx < Cch * 4 * Dch; idx += 256) {
        const int c = idx / (4 * Dch);
        const int j = idx % (4 * Dch);
        float s = 0.f;
        for (int d = 0; d < Dch; ++d) s += Wo[c * Dch + d] * Wf[d * (4 * Dch) + j];
        Mf[idx] = s;
    }
}

// ---------------------------------------------------------------------------
// K1: x = Wi @ (F_in + alpha*prior) + bi    via V_WMMA_F32_16X16X4_F32
//     per block: 16-pixel N-tile, 8 waves = 8 M-tiles (D=128)
// ---------------------------------------------------------------------------
__global__ void k1_ingemm(const float* __restrict__ F_in, const float* __restrict__ prior,
                          const float* __restrict__ Wi, const float* __restrict__ bi,
                          const float* __restrict__ alpha_p,
                          float* __restrict__ xbuf) {
    __shared__ float Bt[Cch * 16];             // 64 x 16 B-panel (4 KB)
    const int tid = threadIdx.x;               // 256
    const int b    = (int)(blockIdx.x / (HW / 16));
    const int pix0 = (int)(blockIdx.x % (HW / 16)) * 16;
    const float alpha = alpha_p[0];

    for (int r = tid; r < Cch * 16; r += 256) {
        const int c = r >> 4, n = r & 15;
        Bt[r] = F_in[((size_t)b * Cch + c) * HW + pix0 + n]
              + alpha * prior[(size_t)b * HW + pix0 + n];
    }
    __syncthreads();

    const int wave = tid >> 5, lane = tid & 31;
    const int m = lane & 15;                   // A row within tile
    const int n = lane & 15;                   // B/D column
    const int ksel = (lane >> 4) << 1;         // lanes 0-15 -> K=(0,1); 16-31 -> K=(2,3)
    const float* Arow = Wi + (size_t)(wave * 16 + m) * Cch;

    v8f acc = {0.f, 0.f, 0.f, 0.f, 0.f, 0.f, 0.f, 0.f};
    for (int k = 0; k < Cch; k += 4) {
        v2f a, bv;
        a.x  = Arow[k + ksel];
        a.y  = Arow[k + ksel + 1];
        bv.x = Bt[(k + ksel) * 16 + n];
        bv.y = Bt[(k + ksel + 1) * 16 + n];
        acc = __builtin_amdgcn_wmma_f32_16x16x4_f32(false, a, false, bv,
                                                    (short)0, acc, false, false);
    }

    const int rsel = (lane < 16) ? 0 : 8;      // C/D layout: VGPR i = rows i / i+8
    for (int i = 0; i < 8; ++i) {
        const int d = wave * 16 + i + rsel;
        xbuf[((size_t)b * Dch + d) * HW + pix0 + n] = acc[i] + bi[d];
    }
}

// ---------------------------------------------------------------------------
// K2: all four directional scans for one (b,d) 128x128 slab, staged in LDS.
//     h_i = A*h_{i-1} + Bc*x_i ; padded stride -> bank-conflict-free both ways.
// ---------------------------------------------------------------------------
__global__ void k2_scan(const float* __restrict__ xbuf, const float* __restrict__ At,
                        const float* __restrict__ Bparam,
                        float* __restrict__ h_lr, float* __restrict__ h_rl,
                        float* __restrict__ h_tb, float* __restrict__ h_bt) {
    __shared__ float xt[Him * (Wim + 1)];      // 66 KB
    __shared__ float ot[Him * (Wim + 1)];      // 66 KB
    const int t  = threadIdx.x;                // 128
    const int bd = blockIdx.x;                 // b*Dch + d
    const int d  = bd & (Dch - 1);
    const size_t base = (size_t)bd * HW;

    for (int j = 0; j < Him; ++j)
        xt[j * (Wim + 1) + t] = xbuf[base + j * Wim + t];
    __syncthreads();

    const float A = At[d], Bc = Bparam[d];
    float h;

    // left -> right (along W, thread t owns row h=t)
    h = 0.f;
    for (int i = 0; i < Wim; ++i) { h = A * h + Bc * xt[t * (Wim + 1) + i]; ot[t * (Wim + 1) + i] = h; }
    __syncthreads();
    for (int j = 0; j < Him; ++j) h_lr[base + j * Wim + t] = ot[j * (Wim + 1) + t];
    __syncthreads();

    // right -> left
    h = 0.f;
    for (int i = Wim - 1; i >= 0; --i) { h = A * h + Bc * xt[t * (Wim + 1) + i]; ot[t * (Wim + 1) + i] = h; }
    __syncthreads();
    for (int j = 0; j < Him; ++j) h_rl[base + j * Wim + t] = ot[j * (Wim + 1) + t];
    __syncthreads();

    // top -> bottom (along H, thread t owns column w=t)
    h = 0.f;
    for (int j = 0; j < Him; ++j) { h = A * h + Bc * xt[j * (Wim + 1) + t]; ot[j * (Wim + 1) + t] = h; }
    __syncthreads();
    for (int j = 0; j < Him; ++j) h_tb[base + j * Wim + t] = ot[j * (Wim + 1) + t];
    __syncthreads();

    // bottom -> top
    h = 0.f;
    for (int j = Him - 1; j >= 0; --j) { h = A * h + Bc * xt[j * (Wim + 1) + t]; ot[j * (Wim + 1) + t] = h; }
    __syncthreads();
    for (int j = 0; j < Him; ++j) h_bt[base + j * Wim + t] = ot[j * (Wim + 1) + t];
}

// ---------------------------------------------------------------------------
// K4: out = F_mod + gamma * (M @ h4 + c0)   via V_WMMA_F32_16X16X4_F32
//     per block: 64-pixel N-tile (4 x 16-pixel panels), 16 waves:
//       wave = panel(ng: 0..3) x C-tile(mt: 0..3), K = 512.
//     Panels strided by 8192+16 floats -> each panel in a distinct 16-bank
//     window: conflict-free coop stores and fragment loads.
// ---------------------------------------------------------------------------
#define PIXT 64
#define PSTRIDE (4 * Dch * 16 + 16)            // 8208 floats per panel

__global__ __launch_bounds__(512)
void k4_outgemm(const float* __restrict__ Mf, const float* __restrict__ c0,
                const float* __restrict__ h4,
                const float* __restrict__ F_in, const float* __restrict__ prior,
                const float* __restrict__ alpha_p, const float* __restrict__ gamma_p,
                float* __restrict__ out) {
    __shared__ float Bt[4 * PSTRIDE];          // ~128 KB
    const int tid     = threadIdx.x;           // 512
    const int b       = (int)(blockIdx.x / (HW / PIXT));
    const int pixbase = (int)(blockIdx.x % (HW / PIXT)) * PIXT;

    // cooperative load: 512 rows x 64 pixels, 256B contiguous per global row
    for (int r = tid; r < 4 * Dch * PIXT; r += 512) {
        const int j  = r >> 6;                 // 0..511 = dir*128 + d
        const int nn = r & 63;                 // pixel within block tile
        const float* rowp = h4 + (size_t)(j >> 7) * BDHW
                               + ((size_t)b * Dch + (j & 127)) * HW;
        Bt[(nn >> 4) * PSTRIDE + j * 16 + (nn & 15)] = rowp[pixbase + nn];
    }
    __syncthreads();

    const int wave = tid >> 5, lane = tid & 31;
    const int ng = wave >> 2;                  // pixel sub-tile 0..3
    const int mt = wave & 3;                   // C tile 0..3
    const int m = lane & 15, n = lane & 15;
    const int ksel = (lane >> 4) << 1;         // lanes 0-15 -> K=(0,1); 16-31 -> K=(2,3)
    const float* Arow = Mf + (size_t)(mt * 16 + m) * (4 * Dch);
    const float* Pan  = Bt + ng * PSTRIDE;

    v8f acc = {0.f, 0.f, 0.f, 0.f, 0.f, 0.f, 0.f, 0.f};
    for (int k = 0; k < 4 * Dch; k += 4) {
        v2f a, bv;
        a.x  = Arow[k + ksel];
        a.y  = Arow[k + ksel + 1];
        bv.x = Pan[(k + ksel) * 16 + n];
        bv.y = Pan[(k + ksel + 1) * 16 + n];
        acc = __builtin_amdgcn_wmma_f32_16x16x4_f32(false, a, false, bv,
                                                    (short)0, acc, false, false);
    }

    const float alpha = alpha_p[0], gamma = gamma_p[0];
    const int pix0 = pixbase + ng * 16;
    const float pr = prior[(size_t)b * HW + pix0 + n];
    const int rsel = (lane < 16) ? 0 : 8;
    for (int i = 0; i < 8; ++i) {
        const int c = mt * 16 + i + rsel;
        const size_t o = ((size_t)b * Cch + c) * HW + pix0 + n;
        out[o] = F_in[o] + alpha * pr + gamma * (acc[i] + c0[c]);
    }
}

// ---------------------------------------------------------------------------
extern "C" void kernel_launch(void* const* d_in, const int* in_sizes, int n_in,
                              void* d_out, int out_size, void* d_ws, size_t ws_size,
                              hipStream_t stream) {
    const float* F_in  = (const float*)d_in[0];
    const float* prior = (const float*)d_in[1];
    const float* Wi    = (const float*)d_in[2];
    const float* bi    = (const float*)d_in[3];
    const float* A_p   = (const float*)d_in[4];
    const float* B_p   = (const float*)d_in[5];
    const float* alpha = (const float*)d_in[6];
    const float* gamma = (const float*)d_in[7];
    const float* Wf    = (const float*)d_in[8];
    const float* bf    = (const float*)d_in[9];
    const float* Wo    = (const float*)d_in[10];
    const float* bo    = (const float*)d_in[11];
    float* out = (float*)d_out;

    // workspace layout (floats): x | h4(4 fields) | M | c0 | At
    float* ws   = (float*)d_ws;
    float* xbuf = ws;
    float* h4   = xbuf + BDHW;
    float* Mf   = h4 + 4 * BDHW;
    float* c0b  = Mf + (size_t)Cch * 4 * Dch;
    float* Atb  = c0b + Cch;

    k0_precompute<<<dim3(1), dim3(256), 0, stream>>>(Wo, Wf, bf, bo, A_p, Mf, c0b, Atb);

    k1_ingemm<<<dim3(Bsz * HW / 16), dim3(256), 0, stream>>>(F_in, prior, Wi, bi, alpha, xbuf);

    k2_scan<<<dim3(Bsz * Dch), dim3(128), 0, stream>>>(xbuf, Atb, B_p,
                                                       h4 + 0 * BDHW, h4 + 1 * BDHW,
                                                       h4 + 2 * BDHW, h4 + 3 * BDHW);

    k4_outgemm<<<dim3(Bsz * HW / PIXT), dim3(512), 0, stream>>>(Mf, c0b, h4, F_in, prior,
                                                                alpha, gamma, out);
}